// M2TTSModel_81518479278700
// MI455X (gfx1250) — compile-verified
//
#include <hip/hip_runtime.h>

// Problem shape (fixed by the reference): B=16, S=512, H=384, T=max_length=4096.
#define B_      16
#define S_      512
#define H_      384
#define T_MAX   4096

typedef float f4 __attribute__((ext_vector_type(4)));

// Address-space-qualified pointer types for the gfx1250 async-LDS builtin.
typedef __attribute__((address_space(1))) int gas_int;   // global
typedef __attribute__((address_space(3))) int las_int;   // LDS

// ---------------------------------------------------------------------------
// Kernel 1: per-batch inclusive scan of durations -> ends[b][s], totals[b].
// One block per batch, 512 threads, Hillis-Steele scan in LDS (9 steps).
// ---------------------------------------------------------------------------
__global__ __launch_bounds__(S_) void lr_scan_durations(const int* __restrict__ dur,
                                                        int* __restrict__ ends,
                                                        int* __restrict__ totals) {
    __shared__ int buf[S_];
    const int b = blockIdx.x;
    const int i = threadIdx.x;
    buf[i] = dur[b * S_ + i];
    __syncthreads();
#pragma unroll
    for (int off = 1; off < S_; off <<= 1) {
        int add = (i >= off) ? buf[i - off] : 0;
        int v   = buf[i];
        __syncthreads();
        buf[i] = v + add;
        __syncthreads();
    }
    ends[b * S_ + i] = buf[i];
    if (i == S_ - 1) totals[b] = buf[i];
}

// ---------------------------------------------------------------------------
// Kernel 2: expand. grid = (T_MAX/ROWS_PER_BLOCK, B). 256 threads = 8 waves.
// Each wave32 handles one output row at a time: 32 lanes x 3 float4 = 1536 B
// = exactly one H=384 f32 row. ends[] staged in LDS (async path on gfx1250),
// binary search is wave-uniform -> LDS broadcast reads. Output stores are
// non-temporal (write-once 100 MB stream); encoder reads stay cacheable
// (12.6 MB tensor is L2-resident and heavily reused).
// ---------------------------------------------------------------------------
#define WAVES_PER_BLOCK 8
#define ROWS_PER_WAVE   4
#define ROWS_PER_BLOCK  (WAVES_PER_BLOCK * ROWS_PER_WAVE)   // 32

__global__ __launch_bounds__(256) void lr_expand(const float* __restrict__ enc,
                                                 const int* __restrict__ ends,
                                                 const int* __restrict__ totals,
                                                 float* __restrict__ out) {
    __shared__ __align__(16) int sEnds[S_];
    const int b   = blockIdx.y;
    const int tid = threadIdx.x;

    // ---- Stage ends[b][0..511] (2 KB) into LDS ----
#if __has_builtin(__builtin_amdgcn_global_load_async_to_lds_b32) && \
    __has_builtin(__builtin_amdgcn_s_wait_asynccnt)
    {
        const int* gsrc = ends + b * S_;
        for (int i = tid; i < S_; i += 256) {
            __builtin_amdgcn_global_load_async_to_lds_b32(
                (gas_int*)(gsrc + i),
                (las_int*)(&sEnds[i]),
                /*offset=*/0, /*cpol=*/0);
        }
        __builtin_amdgcn_s_wait_asynccnt(0);
    }
#else
    for (int i = tid; i < S_; i += 256) sEnds[i] = ends[b * S_ + i];
#endif
    const int total = totals[b];
    __syncthreads();

    const int lane = tid & 31;
    const int wave = tid >> 5;
    const int t0   = blockIdx.x * ROWS_PER_BLOCK + wave * ROWS_PER_WAVE;

#pragma unroll
    for (int r = 0; r < ROWS_PER_WAVE; ++r) {
        const int t = t0 + r;
        f4* dst = (f4*)(out + ((size_t)b * T_MAX + t) * H_);
        if (t >= total) {
            f4 z = (f4){0.f, 0.f, 0.f, 0.f};
#pragma unroll
            for (int k = 0; k < 3; ++k)
                __builtin_nontemporal_store(z, dst + lane + 32 * k);
        } else {
            // searchsorted(ends, t, side='right'): first idx with ends[idx] > t.
            int lo = 0, hi = S_;
            while (lo < hi) {
                int mid = (lo + hi) >> 1;
                if (sEnds[mid] <= t) lo = mid + 1; else hi = mid;
            }
            const int idx = (lo < S_ - 1) ? lo : (S_ - 1);
            const f4* src = (const f4*)(enc + ((size_t)b * S_ + idx) * H_);
#pragma unroll
            for (int k = 0; k < 3; ++k) {
                f4 v = src[lane + 32 * k];
                __builtin_nontemporal_store(v, dst + lane + 32 * k);
            }
        }
    }
}

// ---------------------------------------------------------------------------
// Host launcher. d_in[0] = encoder_output (f32, B*S*H), d_in[1] = durations
// (int, B*S), d_in[2] = max_length scalar (4096, fixed by reference shapes).
// d_ws: B*S ints for ends + B ints for totals (~33 KB).
// ---------------------------------------------------------------------------
extern "C" void kernel_launch(void* const* d_in, const int* in_sizes, int n_in,
                              void* d_out, int out_size, void* d_ws, size_t ws_size,
                              hipStream_t stream) {
    (void)in_sizes; (void)n_in; (void)out_size; (void)ws_size;
    const float* enc = (const float*)d_in[0];
    const int*   dur = (const int*)d_in[1];

    int* ends   = (int*)d_ws;
    int* totals = ends + B_ * S_;

    lr_scan_durations<<<B_, S_, 0, stream>>>(dur, ends, totals);

    dim3 grid(T_MAX / ROWS_PER_BLOCK, B_);
    lr_expand<<<grid, 256, 0, stream>>>(enc, ends, totals, (float*)d_out);
}